// AttentiveTransformer_89859305767532
// MI455X (gfx1250) — compile-verified
//
#include <hip/hip_runtime.h>
#include <hip/hip_bf16.h>

// Problem constants (match reference)
#define B_ROWS 262144
#define A_DIM  64
#define F_DIM  128
#define WT_STRIDE 68      // dword stride for transposed W' in LDS: conflict-free, 8B aligned
#define EPS_BN 1e-5f
#define EPS_LOG 1e-10f
#define SPARSITY_COEF 1e-5f

typedef __attribute__((ext_vector_type(2))) float v2f;
typedef __attribute__((ext_vector_type(8))) float v8f;

__global__ __launch_bounds__(256) void attentive_sparsemax_kernel(
    const float* __restrict__ X,      // [B, 64]
    const float* __restrict__ prior,  // [B, 128]
    const float* __restrict__ W,      // [64, 128]
    const float* __restrict__ bvec,   // [128]
    const float* __restrict__ gamma,  // [128]
    const float* __restrict__ beta,   // [128]
    const float* __restrict__ rmean,  // [128]
    const float* __restrict__ rvar,   // [128]
    float* __restrict__ out,          // [B, 128]
    float* __restrict__ loss)         // scalar accumulator (pre-zeroed)
{
    __shared__ float sWt[F_DIM * WT_STRIDE]; // W' transposed: sWt[f*68 + a]
    __shared__ float sScale[F_DIM];
    __shared__ float sBias[F_DIM];

    const int tid = threadIdx.x;

    // Fold BN into per-column scale/bias
    if (tid < F_DIM) {
        float sc = gamma[tid] * rsqrtf(rvar[tid] + EPS_BN);
        sScale[tid] = sc;
        sBias[tid]  = (bvec[tid] - rmean[tid]) * sc + beta[tid];
    }
    __syncthreads();

    // Stage W' (BN-scaled, transposed) into LDS; coalesced global reads
    for (int i = tid; i < A_DIM * F_DIM; i += 256) {
        int a = i >> 7;           // 0..63
        int f = i & (F_DIM - 1);  // 0..127
        sWt[f * WT_STRIDE + a] = W[i] * sScale[f];
    }
    __syncthreads();

    const int wave = tid >> 5;           // 0..7, each wave owns 16 rows
    const int lane = tid & 31;
    const int half = lane >> 4;          // 0 or 1
    const int n16  = lane & 15;          // 0..15
    const long long m0 = (long long)blockIdx.x * 128 + (long long)wave * 16;

    // Accumulators initialized with folded bias: C layout -> col = 16*nt + n16
    v8f acc[8];
#pragma unroll
    for (int nt = 0; nt < 8; ++nt) {
        float bb = sBias[nt * 16 + n16];
        v8f c = {bb, bb, bb, bb, bb, bb, bb, bb};
        acc[nt] = c;
    }

    // A fragment row for this lane: M = n16 (lanes 0-15 and 16-31 both map M=0..15)
    const float* __restrict__ arow = X + (m0 + n16) * A_DIM;

    // GEMM: K=64 as 16 chained V_WMMA_F32_16X16X4_F32 steps
#pragma unroll
    for (int kk = 0; kk < 16; ++kk) {
        const int kb = kk * 4 + half * 2;   // A layout: v0=K(kb), v1=K(kb+1)
        v2f afrag;
        {
            float2 t = *reinterpret_cast<const float2*>(arow + kb); // 8B aligned
            afrag.x = t.x; afrag.y = t.y;
        }
#pragma unroll
        for (int nt = 0; nt < 8; ++nt) {
            const float* wp = &sWt[(nt * 16 + n16) * WT_STRIDE + kb];
            v2f bfrag;
            {
                float2 t = *reinterpret_cast<const float2*>(wp);    // ds_load_b64
                bfrag.x = t.x; bfrag.y = t.y;
            }
            acc[nt] = __builtin_amdgcn_wmma_f32_16x16x4_f32(
                false, afrag, false, bfrag, (short)0, acc[nt], false, false);
        }
    }

    // Per-row: prior scale, sparsemax via bisection on tau, store + loss partial.
    // C/D layout: VGPR r holds M = r + 8*half, N = 16*nt + n16.
    float lsum = 0.0f;
#pragma unroll
    for (int r = 0; r < 8; ++r) {
        const long long m = m0 + r + half * 8;
        const float* __restrict__ prow = prior + m * F_DIM;
        float* __restrict__ orow = out + m * F_DIM;

        float zr[8];
        float mx = -3.402823466e38f;
#pragma unroll
        for (int nt = 0; nt < 8; ++nt) {
            float z = acc[nt][r] * prow[nt * 16 + n16];
            zr[nt] = z;
            mx = fmaxf(mx, z);
        }
        // row max across the 16 lanes of this half (xor masks < 16 stay in half)
#pragma unroll
        for (int mask = 1; mask <= 8; mask <<= 1)
            mx = fmaxf(mx, __shfl_xor(mx, mask, 32));

        // sum(relu(z - (mx-1))) >= 1 and sum(relu(z - mx)) == 0  =>  tau in [mx-1, mx]
        float lo = mx - 1.0f, hi = mx;
        for (int it = 0; it < 24; ++it) {
            float mid = 0.5f * (lo + hi);
            float s = 0.0f;
#pragma unroll
            for (int nt = 0; nt < 8; ++nt)
                s += fmaxf(zr[nt] - mid, 0.0f);
#pragma unroll
            for (int mask = 1; mask <= 8; mask <<= 1)
                s += __shfl_xor(s, mask, 32);
            bool ge = (s >= 1.0f);
            lo = ge ? mid : lo;   // branch-free: v_cndmask
            hi = ge ? hi  : mid;
        }
        const float tau = 0.5f * (lo + hi);

#pragma unroll
        for (int nt = 0; nt < 8; ++nt) {
            float p = fmaxf(zr[nt] - tau, 0.0f);
            orow[nt * 16 + n16] = p;
            lsum += p * __logf(p + EPS_LOG);
        }
    }

    // Full-wave reduce of loss partial, one atomic per wave
#pragma unroll
    for (int mask = 1; mask <= 16; mask <<= 1)
        lsum += __shfl_xor(lsum, mask, 32);
    if (lane == 0)
        atomicAdd(loss, lsum * (SPARSITY_COEF / (float)B_ROWS));
}

extern "C" void kernel_launch(void* const* d_in, const int* in_sizes, int n_in,
                              void* d_out, int out_size, void* d_ws, size_t ws_size,
                              hipStream_t stream) {
    (void)in_sizes; (void)n_in; (void)d_ws; (void)ws_size; (void)out_size;
    const float* X     = (const float*)d_in[0];
    const float* prior = (const float*)d_in[1];
    const float* W     = (const float*)d_in[2];
    const float* bvec  = (const float*)d_in[3];
    const float* gamma = (const float*)d_in[4];
    const float* beta  = (const float*)d_in[5];
    const float* rmean = (const float*)d_in[6];
    const float* rvar  = (const float*)d_in[7];

    float* out  = (float*)d_out;
    float* loss = out + (size_t)B_ROWS * F_DIM;

    // Zero the scalar loss accumulator (graph-capture safe)
    hipMemsetAsync(loss, 0, sizeof(float), stream);

    dim3 grid(B_ROWS / 128);   // 2048 blocks; each block = 8 waves x 16 rows
    dim3 block(256);
    attentive_sparsemax_kernel<<<grid, block, 0, stream>>>(
        X, prior, W, bvec, gamma, beta, rmean, rvar, out, loss);
}